// RailGNN_86741159510435
// MI455X (gfx1250) — compile-verified
//
#include <hip/hip_runtime.h>

#define N_NODES_C  500000
#define N_EDGES_C  20000000
#define HID        32
#define WAVES_PB   4        // waves per block in MLP kernel
#define LDS_ROW    33       // padded row (floats) to avoid LDS bank conflicts

typedef float v2f __attribute__((ext_vector_type(2)));
typedef float v8f __attribute__((ext_vector_type(8)));

// ---------------------------------------------------------------------------
// Kernel 1: zero the workspace (nbr_sum f32[500000*4] ++ deg u32[500000])
// ---------------------------------------------------------------------------
__global__ void railgnn_zero_ws(unsigned int* __restrict__ ws, int nwords) {
    int i = blockIdx.x * blockDim.x + threadIdx.x;
    if (i < nwords) ws[i] = 0u;
}

// ---------------------------------------------------------------------------
// Kernel 2: edge aggregation. 320MB of int64 index streaming (HBM-bound);
// gather of x rows and the scattered atomics stay resident in the 192MB L2.
// ---------------------------------------------------------------------------
__global__ __launch_bounds__(256)
void railgnn_edge_agg(const long long* __restrict__ ei,   // (2, E) row-major
                      const float*     __restrict__ x,    // (N, 4)
                      float*           __restrict__ nbr,  // (N, 4) sums
                      unsigned int*    __restrict__ deg)  // (N,)
{
    int e = blockIdx.x * blockDim.x + threadIdx.x;
    if (e >= N_EDGES_C) return;
    int src = (int)ei[e];
    int dst = (int)ei[(long long)N_EDGES_C + e];
    const float4* x4 = reinterpret_cast<const float4*>(x);
    float4 v = x4[src];
    float* base = nbr + (long long)dst * 4;
    // native global_atomic_add_f32 (no CAS loop)
    unsafeAtomicAdd(base + 0, v.x);
    unsafeAtomicAdd(base + 1, v.y);
    unsafeAtomicAdd(base + 2, v.z);
    unsafeAtomicAdd(base + 3, v.w);
    atomicAdd(deg + dst, 1u);
}

// ---------------------------------------------------------------------------
// Kernel 3: fused aggregate-combine + 3-layer MLP using V_WMMA_F32_16X16X4_F32.
// One wave32 per 16-node tile. All WMMA branches are wave-uniform (EXEC==all 1s).
// ---------------------------------------------------------------------------
__device__ __forceinline__ v8f wmma_k4(v2f a, v2f b, v8f c) {
    // (neg_a, A, neg_b, B, c_mod, C, reuse_a, reuse_b)
    return __builtin_amdgcn_wmma_f32_16x16x4_f32(false, a, false, b, (short)0, c,
                                                 false, false);
}

__global__ __launch_bounds__(32 * WAVES_PB)
void railgnn_mlp(const float* __restrict__ x,
                 const float* __restrict__ nbr,
                 const unsigned int* __restrict__ deg,
                 const float* __restrict__ W1, const float* __restrict__ b1,
                 const float* __restrict__ W2, const float* __restrict__ b2,
                 const float* __restrict__ W3, const float* __restrict__ b3,
                 float* __restrict__ out)
{
    __shared__ float lds[WAVES_PB][16 * LDS_ROW];

    const int lane = threadIdx.x & 31;
    const int wave = threadIdx.x >> 5;
    const int tile = blockIdx.x * WAVES_PB + wave;
    const int nTiles = N_NODES_C / 16;        // 31250, exact
    if (tile >= nTiles) return;               // wave-uniform

    const int m  = lane & 15;                 // row (A) / column-within-half (B,C)
    const int hi = lane >> 4;                 // 0: lanes 0-15, 1: lanes 16-31
    const int k0 = hi * 2;                    // A/B K-pair owned by this half
    const int node = tile * 16 + m;

    // ---- layer 0: aggregate-combine, loaded directly in 16x4 A-layout ----
    // A layout: VGPR0 = K {0|2}, VGPR1 = K {1|3} for row M = lane%16.
    float xv0 = x  [node * 4 + k0];
    float xv1 = x  [node * 4 + k0 + 1];
    float sv0 = nbr[node * 4 + k0];
    float sv1 = nbr[node * 4 + k0 + 1];
    unsigned int d = deg[node];
    float invd = 1.0f / fmaxf((float)d, 1.0f);
    v2f a0;
    a0.x = d ? (xv0 + sv0 * invd) * 0.5f : xv0;
    a0.y = d ? (xv1 + sv1 * invd) * 0.5f : xv1;

    // ---- layer 1: h0(16x4) @ W1(4x32) + b1, two N-halves ----
    const int n0 = m, n1 = 16 + m;
    v8f acc0, acc1;
    {
        v2f bB0, bB1;   // B 4x16 layout mirrors A: V0 holds K {0|2}, V1 holds K {1|3}
        bB0.x = W1[(k0    ) * HID + n0];  bB0.y = W1[(k0 + 1) * HID + n0];
        bB1.x = W1[(k0    ) * HID + n1];  bB1.y = W1[(k0 + 1) * HID + n1];
        float c0 = b1[n0], c1 = b1[n1];
        acc0 = (v8f){c0, c0, c0, c0, c0, c0, c0, c0};
        acc1 = (v8f){c1, c1, c1, c1, c1, c1, c1, c1};
        acc0 = wmma_k4(a0, bB0, acc0);
        acc1 = wmma_k4(a0, bB1, acc1);
    }

    // relu + restage 16x32 tile to LDS (C-layout -> row-major) for A reload
    float* sl = lds[wave];
    #pragma unroll
    for (int j = 0; j < 8; ++j) {
        int row = j + hi * 8;                // C layout: VGPR j = row j + 8*hi
        sl[row * LDS_ROW + m]      = fmaxf((float)acc0[j], 0.0f);
        sl[row * LDS_ROW + 16 + m] = fmaxf((float)acc1[j], 0.0f);
    }
    __asm volatile("" ::: "memory");         // keep DS loads after DS stores

    // ---- layer 2: h1(16x32) @ W2(32x32) + b2, 8 K-slices x 2 N-halves ----
    v8f acc2_0, acc2_1;
    {
        float c0 = b2[n0], c1 = b2[n1];
        acc2_0 = (v8f){c0, c0, c0, c0, c0, c0, c0, c0};
        acc2_1 = (v8f){c1, c1, c1, c1, c1, c1, c1, c1};
    }
    #pragma unroll
    for (int kk = 0; kk < 8; ++kk) {
        const int kb = kk * 4 + k0;          // global K of this lane-half's pair
        v2f a;
        a.x = sl[m * LDS_ROW + kb];
        a.y = sl[m * LDS_ROW + kb + 1];
        v2f bB0, bB1;
        bB0.x = W2[(kb    ) * HID + n0];  bB0.y = W2[(kb + 1) * HID + n0];
        bB1.x = W2[(kb    ) * HID + n1];  bB1.y = W2[(kb + 1) * HID + n1];
        acc2_0 = wmma_k4(a, bB0, acc2_0);
        acc2_1 = wmma_k4(a, bB1, acc2_1);
    }

    // ---- layer 3: relu(h2)(16x32) @ W3(32x1) + b3, in-register reduction ----
    float w3a = W3[n0], w3b = W3[n1];
    float bias3 = b3[0];
    float partial[8];
    #pragma unroll
    for (int j = 0; j < 8; ++j) {
        float u0 = fmaxf((float)acc2_0[j], 0.0f);
        float u1 = fmaxf((float)acc2_1[j], 0.0f);
        partial[j] = u0 * w3a + u1 * w3b;    // this lane's two columns
    }
    // sum over the 16 columns held by each lane-half (xor masks stay in-half)
    #pragma unroll
    for (int j = 0; j < 8; ++j) {
        float p = partial[j];
        p += __shfl_xor(p, 1, 32);
        p += __shfl_xor(p, 2, 32);
        p += __shfl_xor(p, 4, 32);
        p += __shfl_xor(p, 8, 32);
        partial[j] = p;
    }
    if (m < 8) {
        float r = partial[0];
        #pragma unroll
        for (int j = 1; j < 8; ++j) r = (m == j) ? partial[j] : r;
        out[tile * 16 + hi * 8 + m] = r + bias3;   // row = hi*8 + m
    }
}

// ---------------------------------------------------------------------------
extern "C" void kernel_launch(void* const* d_in, const int* in_sizes, int n_in,
                              void* d_out, int out_size, void* d_ws, size_t ws_size,
                              hipStream_t stream) {
    const float*     x  = (const float*)d_in[0];
    const long long* ei = (const long long*)d_in[1];   // int64 (2, E)
    const float* W1 = (const float*)d_in[2];
    const float* b1 = (const float*)d_in[3];
    const float* W2 = (const float*)d_in[4];
    const float* b2 = (const float*)d_in[5];
    const float* W3 = (const float*)d_in[6];
    const float* b3 = (const float*)d_in[7];
    float* out = (float*)d_out;

    float*        nbr = (float*)d_ws;                                  // 8 MB
    unsigned int* deg = (unsigned int*)((char*)d_ws + (size_t)N_NODES_C * 4 * sizeof(float));

    // 1) zero accumulators (atomics below require it every call)
    const int nwords = N_NODES_C * 4 + N_NODES_C;
    railgnn_zero_ws<<<(nwords + 255) / 256, 256, 0, stream>>>((unsigned int*)d_ws, nwords);

    // 2) edge aggregation
    railgnn_edge_agg<<<(N_EDGES_C + 255) / 256, 256, 0, stream>>>(ei, x, nbr, deg);

    // 3) combine + WMMA MLP
    const int nTiles = N_NODES_C / 16;                 // 31250
    const int blocks = (nTiles + WAVES_PB - 1) / WAVES_PB;
    railgnn_mlp<<<blocks, 32 * WAVES_PB, 0, stream>>>(x, nbr, deg,
                                                      W1, b1, W2, b2, W3, b3, out);
    (void)in_sizes; (void)n_in; (void)out_size; (void)ws_size;
}